// WaveletTransform_53309134078401
// MI455X (gfx1250) — compile-verified
//
#include <hip/hip_runtime.h>

// 2D Haar DWT, (32,3,512,512) f32 -> (32,3,4,256,256) f32.
// Bandwidth-bound: ~200 MB traffic, ~75 MFLOP. Strategy: CDNA5 async
// global->LDS copies (ASYNCcnt) with per-wave double buffering, VALU
// butterfly, non-temporal b64 stores.

typedef __attribute__((ext_vector_type(4))) float f32x4;
typedef __attribute__((ext_vector_type(2))) float f32x2;

#define WAVES 8
#define ITERS 4

// Copy one 4KB row-pair tile (2 rows x 512 f32) from global to LDS.
// Lane supplies base+lane*16; INST_OFFSET advances both global and LDS
// addresses (CDNA5 ISA), so 8 b128 ops cover 4KB.
__device__ __forceinline__ void async_tile_load(const float* g, unsigned lds) {
  asm volatile(
      "global_load_async_to_lds_b128 %0, %1, off\n\t"
      "global_load_async_to_lds_b128 %0, %1, off offset:512\n\t"
      "global_load_async_to_lds_b128 %0, %1, off offset:1024\n\t"
      "global_load_async_to_lds_b128 %0, %1, off offset:1536\n\t"
      "global_load_async_to_lds_b128 %0, %1, off offset:2048\n\t"
      "global_load_async_to_lds_b128 %0, %1, off offset:2560\n\t"
      "global_load_async_to_lds_b128 %0, %1, off offset:3072\n\t"
      "global_load_async_to_lds_b128 %0, %1, off offset:3584"
      :
      : "v"(lds), "v"(g)
      : "memory");
}

__global__ __launch_bounds__(WAVES * 32)
void haar2d_async_kernel(const float* __restrict__ x, float* __restrict__ out,
                         int nrp) {
  // Per-wave double buffer: 2 x (2 rows x 512 f32) = 8KB/wave, 64KB/block.
  __shared__ __align__(16) float tile[WAVES * 2048];

  const int lane = threadIdx.x & 31;
  const int wave = threadIdx.x >> 5;

  const int rpBase = ((int)blockIdx.x * WAVES + wave) * ITERS;
  if (rpBase >= nrp) return;

  float* buf0 = &tile[wave * 2048];
  float* buf1 = buf0 + 1024;
  // Low 32 bits of a generic pointer into LDS == LDS byte offset
  // (shared aperture base has zero low 32 bits; CDNA5 ISA).
  const unsigned lds0 = (unsigned)(size_t)buf0 + (unsigned)lane * 16u;
  const unsigned lds1 = (unsigned)(size_t)buf1 + (unsigned)lane * 16u;

  // Input row-pair rp occupies 1024 consecutive floats (2 rows of 512).
  const float* g = x + (size_t)rpBase * 1024 + (size_t)lane * 4;

  async_tile_load(g, lds0);  // prime the pipeline into buffer 0

  for (int it = 0; it < ITERS; ++it) {
    const int rp = rpBase + it;
    if (rp >= nrp) break;

    const bool pf = (it + 1 < ITERS) && (rp + 1 < nrp);
    if (pf) {
      // Prefetch next tile into the other buffer, then wait for the
      // current tile only (async loads complete in order: <=8 left
      // outstanding means the first batch of 8 has landed).
      async_tile_load(g + (size_t)(it + 1) * 1024, (it & 1) ? lds0 : lds1);
      asm volatile("s_wait_asynccnt 8" ::: "memory");
    } else {
      asm volatile("s_wait_asynccnt 0" ::: "memory");
    }

    const float* row0 = (it & 1) ? buf1 : buf0;
    const float* row1 = row0 + 512;

    const int plane = rp >> 8;   // b*3 + c
    const int h     = rp & 255;  // output row
    // out[plane][s][h][:], subband plane stride = 256*256 = 65536
    float* ob = out + (size_t)plane * 262144 + (size_t)h * 256 + lane * 2;

#pragma unroll
    for (int k = 0; k < 4; ++k) {
      // Lane reads 4 input cols (2 output blocks) per chunk; dense
      // 16B/lane ds_load_b128 pattern, conflict-free.
      f32x4 r0 = *(const f32x4*)(row0 + k * 128 + lane * 4);
      f32x4 r1 = *(const f32x4*)(row1 + k * 128 + lane * 4);

      // Haar butterfly: 8 adds + 4 muls per 2x2 block.
      float s0 = r0.x + r0.y, d0 = r0.x - r0.y;
      float s1 = r1.x + r1.y, d1 = r1.x - r1.y;
      float t0 = r0.z + r0.w, e0 = r0.z - r0.w;
      float t1 = r1.z + r1.w, e1 = r1.z - r1.w;

      f32x2 vA = {0.5f * (s0 + s1), 0.5f * (t0 + t1)};
      f32x2 vH = {0.5f * (s0 - s1), 0.5f * (t0 - t1)};
      f32x2 vV = {0.5f * (d0 + d1), 0.5f * (e0 + e1)};
      f32x2 vD = {0.5f * (d0 - d1), 0.5f * (e0 - e1)};

      // Dense 8B/lane stores (256B/wave/instr), write-once -> NT hint.
      float* oc = ob + k * 64;
      __builtin_nontemporal_store(vA, (f32x2*)(oc));
      __builtin_nontemporal_store(vH, (f32x2*)(oc + 65536));
      __builtin_nontemporal_store(vV, (f32x2*)(oc + 131072));
      __builtin_nontemporal_store(vD, (f32x2*)(oc + 196608));
    }
  }
}

extern "C" void kernel_launch(void* const* d_in, const int* in_sizes, int n_in,
                              void* d_out, int out_size, void* d_ws,
                              size_t ws_size, hipStream_t stream) {
  const float* x = (const float*)d_in[0];
  float* out = (float*)d_out;

  // Row-pair tiles: each is 2 rows x 512 floats = 1024 floats.
  const int nrp = in_sizes[0] / 1024;  // 32*3*256 = 24576
  const int tiles_per_block = WAVES * ITERS;
  const int blocks = (nrp + tiles_per_block - 1) / tiles_per_block;  // 768

  haar2d_async_kernel<<<dim3(blocks), dim3(WAVES * 32), 0, stream>>>(x, out,
                                                                     nrp);
}